// Solver_74861279969334
// MI455X (gfx1250) — compile-verified
//
#include <hip/hip_runtime.h>
#include <hip/hip_bf16.h>

// Anisotropic eikonal fast-iteration solver on a 1024x1024 triangulated grid.
//   Kernel 1 (once): fold adjacency + vertices + 2x2 tensor field into
//     per-(vertex,slot) {alpha, beta, s0=sqrt(max(gamma,EPS))} (12B, dense
//     b96 loads) + one uint2/vertex holding all twelve 3-bit stencil codes and
//     valid bits.  72MB+8MB tables -> L2-resident (192MB L2).
//   Kernel 2 (x10): Jacobi sweep; per vertex 6x b96 + 1x b64 + 7 stencil
//     u-loads, all index-independent (stencil-code trick) -> full MLP;
//     quadratic solve per slot (5 trans ops), min-reduce, store.
// Bandwidth-bound: ~112 B/vertex/iter -> ~4.8us/iter at 23.3 TB/s HBM,
// faster once coefficient tables are L2-resident.  No dense matmul survives
// the algebraic folding, so WMMA does not apply.  CDNA5 paths used: wave32
// sizing, global_prefetch_b8, dense b96/b64/b128 vector memory.

constexpr int   NG    = 1024;
constexpr int   NV    = NG * NG;          // 1,048,576 vertices
constexpr int   ADJ   = 6;
constexpr float MAXV  = 20.0f;
constexpr int   ITERS = 10;               // even -> ping-pong ends in d_out
constexpr float EPSQ  = 1e-12f;

struct Coef3 { float alpha, beta, s0; };  // s0 = sqrt(max(gamma, EPS))

// Stencil neighbor offsets (all valid j/k deltas by mesh construction):
// code: 0:-(NG+1)  1:-NG  2:-1  3:+1  4:+NG  5:+(NG+1)
__device__ __forceinline__ int code_of(int d) {
  int c = 0;                       // default / -(NG+1)
  c = (d == -NG    ) ? 1 : c;
  c = (d == -1     ) ? 2 : c;
  c = (d ==  1     ) ? 3 : c;
  c = (d ==  NG    ) ? 4 : c;
  c = (d ==  NG + 1) ? 5 : c;
  return c;
}

__device__ __forceinline__ float sel6(int c, const float u[6]) {
  float r = u[0];
  r = (c == 1) ? u[1] : r;
  r = (c == 2) ? u[2] : r;
  r = (c == 3) ? u[3] : r;
  r = (c == 4) ? u[4] : r;
  r = (c == 5) ? u[5] : r;
  return r;
}

__device__ __forceinline__ float clamp01(float x) {
  return fminf(fmaxf(x, 0.0f), 1.0f);
}

// One adjacency slot: min over the 4 lambda candidates of
//   T(lam) = lam*u_j + (1-lam)*u_k + sqrt(max(q(lam), EPS)),
//   q(lam) = gamma - 2*lam*alpha + lam^2*beta   (= d^T M d, M symmetric),
// with gamma = s0^2 and T(0) = u_k + s0 directly.
__device__ __forceinline__ float slot_T(float alpha, float beta, float s0,
                                        float uj, float uk) {
  float gamma  = s0 * s0;
  float delta  = uj - uk;
  float d2     = delta * delta;
  float a      = beta * (beta - d2);
  float b      = 2.0f * alpha * (d2 - beta);
  float c      = alpha * alpha - d2 * gamma;
  float disc   = fmaxf(b * b - 4.0f * a * c, 0.0f);
  float sq     = sqrtf(disc);
  float a_safe = (fabsf(a) > 1e-10f) ? a : 1e-10f;
  float inv2a  = 0.5f / a_safe;
  float lam1   = clamp01((-b + sq) * inv2a);
  float lam2   = clamp01((-b - sq) * inv2a);
  float T0 = uk + s0;                                              // lam = 0
  float T1 = uj + sqrtf(fmaxf(gamma - 2.0f * alpha + beta, EPSQ)); // lam = 1
  float q1 = gamma + lam1 * (lam1 * beta - 2.0f * alpha);
  float q2 = gamma + lam2 * (lam2 * beta - 2.0f * alpha);
  float Ta = lam1 * uj + (1.0f - lam1) * uk + sqrtf(fmaxf(q1, EPSQ));
  float Tb = lam2 * uj + (1.0f - lam2) * uk + sqrtf(fmaxf(q2, EPSQ));
  return fminf(fminf(T0, T1), fminf(Ta, Tb));
}

// ---------------- Kernel 1: fold geometry+tensors into tables -------------
__global__ void __launch_bounds__(256)
precompute_coefs(const float* __restrict__ tf,    // (S,2,2)
                 const float* __restrict__ vts,   // (V,2)
                 const int*   __restrict__ adj,   // (V,6,3)
                 Coef3*       __restrict__ coefs, // [6][V]
                 uint2*       __restrict__ codes) // [V]
{
  int v = blockIdx.x * blockDim.x + threadIdx.x;
  if (v >= NV) return;
  // gfx1250 global_prefetch_b8: pull the 72B adjacency row early
  __builtin_prefetch(&adj[(size_t)v * (ADJ * 3)], 0, 3);

  float xix = vts[2 * v + 0];
  float xiy = vts[2 * v + 1];

  unsigned cx = 0u, cy = 0u;
#pragma unroll
  for (int a = 0; a < ADJ; ++a) {
    int base = v * (ADJ * 3) + a * 3;
    int s = adj[base + 0];
    int j = adj[base + 1];
    int k = adj[base + 2];
    Coef3 c = {0.0f, 0.0f, 0.0f};
    unsigned byte = 0u;
    if (s >= 0) {
      float xjx = vts[2 * j + 0], xjy = vts[2 * j + 1];
      float xkx = vts[2 * k + 0], xky = vts[2 * k + 1];
      float e1x = xix - xkx, e1y = xiy - xky;
      float e2x = xjx - xkx, e2y = xjy - xky;
      float m00 = tf[4 * s + 0], m01 = tf[4 * s + 1];
      float m10 = tf[4 * s + 2], m11 = tf[4 * s + 3];
      float Me1x = m00 * e1x + m01 * e1y;
      float Me1y = m10 * e1x + m11 * e1y;
      float Me2x = m00 * e2x + m01 * e2y;
      float Me2y = m10 * e2x + m11 * e2y;
      float alpha = e2x * Me1x + e2y * Me1y;
      float beta  = e2x * Me2x + e2y * Me2y;
      float gamma = e1x * Me1x + e1y * Me1y;
      c.alpha = alpha;
      c.beta  = beta;
      c.s0    = sqrtf(fmaxf(gamma, EPSQ));
      byte = (unsigned)code_of(j - v) | ((unsigned)code_of(k - v) << 3) | 0x40u;
    }
    coefs[(size_t)a * NV + v] = c;          // dense b96 per lane
    if (a < 4) cx |= byte << (8 * a);
    else       cy |= byte << (8 * (a - 4));
  }
  uint2 cw; cw.x = cx; cw.y = cy;
  codes[v] = cw;
}

// ---------------- init: u0 = MAXV, u0[site] = site value ------------------
__global__ void __launch_bounds__(256)
init_u(float* __restrict__ u, const int* __restrict__ sind,
       const float* __restrict__ sval)
{
  int v = blockIdx.x * blockDim.x + threadIdx.x;
  if (v >= NV) return;
  int   site = sind[0];   // uniform -> scalar load
  float sv   = sval[0];
  u[v] = (v == site) ? sv : MAXV;
}

// ---------------- Kernel 2: one Jacobi sweep (fast, table path) -----------
__global__ void __launch_bounds__(256)
iter_fast(const float* __restrict__ usrc,
          float*       __restrict__ udst,
          const Coef3* __restrict__ coefs,
          const uint2* __restrict__ codes,
          const int*   __restrict__ sind,
          const float* __restrict__ sval)
{
  int v = blockIdx.x * blockDim.x + threadIdx.x;
  if (v >= NV) return;

  // Every load below is index-independent of every other: full MLP.
  // Clamped stencil indices: out-of-range values can only be selected by
  // invalid (masked) slots.
  uint2 cw = codes[v];
  float un[6];
  un[0] = usrc[max(v - NG - 1, 0)];
  un[1] = usrc[max(v - NG,     0)];
  un[2] = usrc[max(v - 1,      0)];
  un[3] = usrc[min(v + 1,      NV - 1)];
  un[4] = usrc[min(v + NG,     NV - 1)];
  un[5] = usrc[min(v + NG + 1, NV - 1)];
  float best = usrc[v];

  Coef3 cf[6];
#pragma unroll
  for (int a = 0; a < ADJ; ++a) {
    cf[a] = coefs[(size_t)a * NV + v];      // 6 dense b96 loads, issued early
  }

#pragma unroll
  for (int a = 0; a < ADJ; ++a) {
    unsigned byte = ((a < 4) ? (cw.x >> (8 * a)) : (cw.y >> (8 * (a - 4)))) & 0xFFu;
    if (byte & 0x40u) {
      float uj = sel6((int)(byte & 7u), un);
      float uk = sel6((int)((byte >> 3) & 7u), un);
      best = fminf(best, slot_T(cf[a].alpha, cf[a].beta, cf[a].s0, uj, uk));
    }
  }
  int   site = sind[0];
  float sv   = sval[0];
  udst[v] = (v == site) ? sv : best;
}

// ---------------- fallback sweep: recompute coefs (small-ws path) ---------
__global__ void __launch_bounds__(256)
iter_slow(const float* __restrict__ usrc,
          float*       __restrict__ udst,
          const float* __restrict__ tf,
          const float* __restrict__ vts,
          const int*   __restrict__ adj,
          const int*   __restrict__ sind,
          const float* __restrict__ sval)
{
  int v = blockIdx.x * blockDim.x + threadIdx.x;
  if (v >= NV) return;
  float xix = vts[2 * v + 0];
  float xiy = vts[2 * v + 1];
  float best = usrc[v];
#pragma unroll
  for (int a = 0; a < ADJ; ++a) {
    int base = v * (ADJ * 3) + a * 3;
    int s = adj[base + 0];
    int j = adj[base + 1];
    int k = adj[base + 2];
    if (s >= 0) {
      float xjx = vts[2 * j + 0], xjy = vts[2 * j + 1];
      float xkx = vts[2 * k + 0], xky = vts[2 * k + 1];
      float e1x = xix - xkx, e1y = xiy - xky;
      float e2x = xjx - xkx, e2y = xjy - xky;
      float m00 = tf[4 * s + 0], m01 = tf[4 * s + 1];
      float m10 = tf[4 * s + 2], m11 = tf[4 * s + 3];
      float Me1x = m00 * e1x + m01 * e1y;
      float Me1y = m10 * e1x + m11 * e1y;
      float Me2x = m00 * e2x + m01 * e2y;
      float Me2y = m10 * e2x + m11 * e2y;
      float alpha = e2x * Me1x + e2y * Me1y;
      float beta  = e2x * Me2x + e2y * Me2y;
      float gamma = e1x * Me1x + e1y * Me1y;
      float s0    = sqrtf(fmaxf(gamma, EPSQ));
      best = fminf(best, slot_T(alpha, beta, s0, usrc[j], usrc[k]));
    }
  }
  int   site = sind[0];
  float sv   = sval[0];
  udst[v] = (v == site) ? sv : best;
}

extern "C" void kernel_launch(void* const* d_in, const int* in_sizes, int n_in,
                              void* d_out, int out_size, void* d_ws, size_t ws_size,
                              hipStream_t stream) {
  const float* tf   = (const float*)d_in[0];  // tensor_field  (S,2,2)
  const float* vts  = (const float*)d_in[1];  // vertices      (V,2)
  const float* sval = (const float*)d_in[2];  // initial_site_values (1,)
  const int*   adj  = (const int*)  d_in[3];  // adjacency_data (V,6,3)
  const int*   sind = (const int*)  d_in[4];  // initial_site_inds (1,)
  float* out = (float*)d_out;

  const size_t uBytes     = (size_t)NV * sizeof(float);        //  4 MB
  const size_t coefBytes  = (size_t)ADJ * NV * sizeof(Coef3);  // 72 MB
  const size_t codesBytes = (size_t)NV * sizeof(uint2);        //  8 MB
  char* ws = (char*)d_ws;

  const bool fast = (ws_size >= coefBytes + codesBytes + uBytes);
  Coef3* coefs = (Coef3*)ws;
  uint2* codes = (uint2*)(ws + coefBytes);
  float* uAlt  = fast ? (float*)(ws + coefBytes + codesBytes) : (float*)ws;

  dim3 blk(256);               // 8 wave32 waves per block
  dim3 grd(NV / 256);          // NV is a multiple of 256

  if (fast) {
    precompute_coefs<<<grd, blk, 0, stream>>>(tf, vts, adj, coefs, codes);
  }
  init_u<<<grd, blk, 0, stream>>>(out, sind, sval);

  // Ping-pong: start in d_out; ITERS is even so the result lands in d_out.
  float* src = out;
  float* dst = uAlt;
  for (int it = 0; it < ITERS; ++it) {
    if (fast) {
      iter_fast<<<grd, blk, 0, stream>>>(src, dst, coefs, codes, sind, sval);
    } else {
      iter_slow<<<grd, blk, 0, stream>>>(src, dst, tf, vts, adj, sind, sval);
    }
    float* t = src; src = dst; dst = t;
  }
}